// BlockAttentionResidual_1726576853595
// MI455X (gfx1250) — compile-verified
//
#include <hip/hip_runtime.h>
#include <cstdint>

// Problem constants (fixed by the reference's setup_inputs)
#define NV     9        // 8 blocks + 1 partial block
#define DD     1024     // feature dim
#define BT     8192     // B*T rows
#define NROWS  10       // 9 V rows + 1 (proj_w*norm_w) row in the A matrix
#define RSTR   1028     // LDS row stride in floats (pad -> conflict-free WMMA reads)

typedef float v2f __attribute__((ext_vector_type(2)));
typedef float v8f __attribute__((ext_vector_type(8)));

__global__ __launch_bounds__(256)
void block_attn_kernel(const float* __restrict__ blocks,
                       const float* __restrict__ partial,
                       const float* __restrict__ proj_w,
                       const float* __restrict__ norm_w,
                       float* __restrict__ out) {
    // Rows 0..8: V rows. Row 9: proj_w*norm_w. Row 10: zeros (pad rows 10..15).
    __shared__ float ldsV[(NROWS + 1) * RSTR];   // 45,232 B
    __shared__ float gbuf[8 * 256];              // 8 per-wave partial 16x16 Grams
    __shared__ float gsum[256];                  // reduced Gram

    const int tid  = threadIdx.x;
    const int bt   = blockIdx.x;           // b*T + t
    const int lane = tid & 31;
    const int w    = tid >> 5;             // wave id 0..7

    // ---------------- Phase 1: async-stage 9 rows (4KB each) into LDS -------
    #pragma unroll
    for (int n = 0; n < NV; ++n) {
        const float* src = (n < 8) ? (blocks + ((size_t)n * BT + bt) * DD)
                                   : (partial + (size_t)bt * DD);
        uint64_t gaddr = (uint64_t)(uintptr_t)(src + tid * 4);
        // Low 32 bits of a generic pointer into __shared__ == LDS byte offset.
        uint32_t laddr = (uint32_t)(uintptr_t)(&ldsV[n * RSTR + tid * 4]);
        asm volatile("global_load_async_to_lds_b128 %0, %1, off"
                     :: "v"(laddr), "v"(gaddr) : "memory");
    }
    // Row 9 of A: proj_w * norm_w (tiny, L2-resident across all 8192 WGs)
    {
        float4 p = ((const float4*)proj_w)[tid];
        float4 g = ((const float4*)norm_w)[tid];
        float4 r; r.x = p.x*g.x; r.y = p.y*g.y; r.z = p.z*g.z; r.w = p.w*g.w;
        *(float4*)&ldsV[9 * RSTR + tid * 4] = r;
    }
    // Row 10: zero pad row (read by lanes whose A-row is 10..15)
    {
        float4 z; z.x = 0.f; z.y = 0.f; z.z = 0.f; z.w = 0.f;
        *(float4*)&ldsV[NROWS * RSTR + tid * 4] = z;
    }
    asm volatile("s_wait_asynccnt 0" ::: "memory");  // my wave's async copies done
    __syncthreads();                                  // everyone's copies visible

    // ---------------- Phase 2: Gram matrix G = A * A^T via WMMA -------------
    // V_WMMA_F32_16X16X4_F32: the A(16x4) and B(4x16) register layouts are
    // mutual transposes (lane l, vgpr v holds A[l%16][v + 2*(l/16)] in both),
    // so feeding the same VGPR pair as A and B yields A*A^T.
    const int row   = lane & 15;            // A-matrix row this lane supplies
    const int khalf = (lane >> 4) << 1;     // lanes 16..31 supply K+2,K+3
    // Loop-invariant per-lane base: rows >= 10 read the zero row (broadcast).
    const float* arow_ptr =
        &ldsV[((row < NROWS) ? row : NROWS) * RSTR + (w << 7) + khalf];
    v8f acc = {};
    #pragma unroll
    for (int kk = 0; kk < 32; ++kk) {       // wave w covers K in [128w,128w+128)
        v2f a = *(const v2f*)(arow_ptr + (kk << 2));   // ds_load_b64, imm offset
        acc = __builtin_amdgcn_wmma_f32_16x16x4_f32(
                  false, a, false, a, (short)0, acc, false, false);
    }
    // Scatter partial Gram to LDS: C/D layout -> vgpr r, lanes 0-15 = M=r,
    // lanes 16-31 = M=r+8, N = lane%16.
    {
        const int c   = lane & 15;
        const int mhi = (lane >> 4) << 3;
        #pragma unroll
        for (int r = 0; r < 8; ++r)
            gbuf[(w << 8) + ((r + mhi) << 4) + c] = acc[r];
    }
    __syncthreads();
    {   // reduce the 8 per-wave Grams
        float s = 0.0f;
        #pragma unroll
        for (int wv = 0; wv < 8; ++wv) s += gbuf[(wv << 8) + tid];
        gsum[tid] = s;
    }
    __syncthreads();

    // ---------------- Phase 3: softmax over n (redundant per thread) --------
    float logit[NV];
    float mx = -3.0e38f;
    #pragma unroll
    for (int n = 0; n < NV; ++n) {
        float ss  = gsum[n * 16 + n];   // sum_d v^2
        float dt  = gsum[n * 16 + 9];   // sum_d v * pw * nw
        float inv = rsqrtf(ss * (1.0f / (float)DD) + 1e-6f);
        logit[n]  = dt * inv;
        mx = fmaxf(mx, logit[n]);
    }
    float wgt[NV];
    float den = 0.0f;
    #pragma unroll
    for (int n = 0; n < NV; ++n) { wgt[n] = __expf(logit[n] - mx); den += wgt[n]; }
    const float rden = 1.0f / den;

    // ---------------- Phase 4: h[d] = sum_n w[n] * V[n][d] ------------------
    const int d0 = tid * 4;
    float4 o; o.x = 0.f; o.y = 0.f; o.z = 0.f; o.w = 0.f;
    #pragma unroll
    for (int n = 0; n < NV; ++n) {
        float4 v = *(const float4*)&ldsV[n * RSTR + d0];
        float wn = wgt[n] * rden;
        o.x = fmaf(wn, v.x, o.x);
        o.y = fmaf(wn, v.y, o.y);
        o.z = fmaf(wn, v.z, o.z);
        o.w = fmaf(wn, v.w, o.w);
    }
    *(float4*)(out + (size_t)bt * DD + d0) = o;
}

extern "C" void kernel_launch(void* const* d_in, const int* in_sizes, int n_in,
                              void* d_out, int out_size, void* d_ws, size_t ws_size,
                              hipStream_t stream) {
    (void)in_sizes; (void)n_in; (void)out_size; (void)d_ws; (void)ws_size;
    const float* blocks  = (const float*)d_in[0];   // [8,4,2048,1024] f32
    const float* partial = (const float*)d_in[1];   // [4,2048,1024]   f32
    const float* proj_w  = (const float*)d_in[2];   // [1024]          f32
    const float* norm_w  = (const float*)d_in[3];   // [1024]          f32
    float* out = (float*)d_out;                     // [4,2048,1024]   f32
    block_attn_kernel<<<BT, 256, 0, stream>>>(blocks, partial, proj_w, norm_w, out);
}